// CharacterGNN_1589137899613
// MI455X (gfx1250) — compile-verified
//
#include <hip/hip_runtime.h>

typedef __attribute__((ext_vector_type(2))) float v2f;
typedef __attribute__((ext_vector_type(4))) float v4f;
typedef __attribute__((ext_vector_type(8))) float v8f;

#define EMB 128
#define HID 256
#define NCLS 64

// ---------------------------------------------------------------------------
// Fold kernel: one block computes
//   M[2][128] = 0.5 * fc_W @ (W2_rel+W2_root) @ (W1_rel+W1_root)
//   c[2]      = fc_W @ ((W2_rel+W2_root) @ b1 + b2) + fc_b
// into d_ws (floats: [0..255]=M row-major, [256..257]=c).
// ---------------------------------------------------------------------------
__global__ __launch_bounds__(256) void gnn_fold_kernel(
    const float* __restrict__ W1_rel, const float* __restrict__ b1,
    const float* __restrict__ W1_root, const float* __restrict__ W2_rel,
    const float* __restrict__ b2, const float* __restrict__ W2_root,
    const float* __restrict__ fc_W, const float* __restrict__ fc_b,
    float* __restrict__ Mc)
{
    __shared__ float T[2 * HID];   // fc_W @ Ws2  -> [2][256]
    __shared__ float u[NCLS];      // Ws2 @ b1 + b2
    const int t = threadIdx.x;

    // T[n][k2] = sum_o fc_W[n][o] * (W2_rel[o][k2] + W2_root[o][k2])
    for (int i = t; i < 2 * HID; i += 256) {
        const int n = i >> 8, k2 = i & (HID - 1);
        float acc = 0.f;
        for (int o = 0; o < NCLS; ++o)
            acc += fc_W[n * NCLS + o] * (W2_rel[o * HID + k2] + W2_root[o * HID + k2]);
        T[i] = acc;
    }
    // u[o] = b2[o] + sum_k Ws2[o][k] * b1[k]
    if (t < NCLS) {
        float acc = b2[t];
        for (int k = 0; k < HID; ++k)
            acc += (W2_rel[t * HID + k] + W2_root[t * HID + k]) * b1[k];
        u[t] = acc;
    }
    __syncthreads();

    // M[n][k] = 0.5 * sum_k2 T[n][k2] * (W1_rel[k2][k] + W1_root[k2][k])
    {
        const int n = t >> 7, k = t & (EMB - 1);
        float acc = 0.f;
        for (int k2 = 0; k2 < HID; ++k2)
            acc += T[n * HID + k2] * (W1_rel[k2 * EMB + k] + W1_root[k2 * EMB + k]);
        Mc[t] = 0.5f * acc;
    }
    if (t < 2) {
        float acc = fc_b[t];
        for (int o = 0; o < NCLS; ++o) acc += fc_W[t * NCLS + o] * u[o];
        Mc[2 * EMB + t] = acc;
    }
}

// ---------------------------------------------------------------------------
// Main kernel: out[b] = M @ (emb[x[b,0]] + emb[x[b,1]]) + c
// One wave32 handles 16 batches via V_WMMA_F32_16X16X4_F32:
//   A (16x4 f32): lane l / l+16 cooperatively hold batch l's summed-embedding
//     K-chunk; B (4x16 f32): M columns (only N=0,1 non-zero, from LDS);
//   C/D (16x16 f32): column n<2, row m -> out[batch m][n]; bias folded into C.
// ---------------------------------------------------------------------------
__global__ __launch_bounds__(256) void gnn_main_kernel(
    const int* __restrict__ x, const float* __restrict__ emb,
    const float* __restrict__ Mc, float* __restrict__ out, int nB)
{
    __shared__ float Bl[4 * EMB];   // rows 0,1 = M ; rows 2,3 = zeros
    const int tid = threadIdx.x;
    for (int i = tid; i < 4 * EMB; i += 256)
        Bl[i] = (i < 2 * EMB) ? Mc[i] : 0.f;
    __syncthreads();

    const int lane = tid & 31;
    const int half = lane >> 4;      // which lane-half (K split)
    const int n    = lane & 15;      // column for B/C/D, batch row for A
    const int wave = (blockIdx.x * 256 + tid) >> 5;
    const int batch_base = wave * 16;
    const int batch = batch_base + n;
    const int bcl = (batch < nB) ? batch : (nB - 1);   // clamp: keep EXEC full

    const int idx0 = x[2 * bcl + 0];
    const int idx1 = x[2 * bcl + 1];
    const float* __restrict__ e0 = emb + (size_t)idx0 * EMB;
    const float* __restrict__ e1 = emb + (size_t)idx1 * EMB;

    const float cv = (n < 2) ? Mc[2 * EMB + n] : 0.f;
    v8f acc;
#pragma unroll
    for (int r = 0; r < 8; ++r) acc[r] = cv;

    const int brow = (n < 2) ? n : 2;                  // n>=2 -> zero row
    const float* __restrict__ brp = Bl + brow * EMB;

#pragma unroll
    for (int kb = 0; kb < EMB; kb += 8) {
        const int off = kb + 4 * half;                 // this half's K chunk
        const v4f f0 = *(const v4f*)(e0 + off);
        const v4f f1 = *(const v4f*)(e1 + off);
        const v4f s  = f0 + f1;                        // summed embedding
        const v4f bm = *(const v4f*)(brp + off);       // matching M values

        v2f a0 = {s.x, s.y}, a1 = {s.z, s.w};
        v2f b0 = {bm.x, bm.y}, b1v = {bm.z, bm.w};
        acc = __builtin_amdgcn_wmma_f32_16x16x4_f32(
            false, a0, false, b0, (short)0, acc, false, false);
        acc = __builtin_amdgcn_wmma_f32_16x16x4_f32(
            false, a1, false, b1v, (short)0, acc, false, false);
    }

    // D layout: VGPR r -> row (half*8 + r), column n = lane&15.
    if (n < 2) {
#pragma unroll
        for (int r = 0; r < 8; ++r) {
            const int bm = batch_base + half * 8 + r;
            if (bm < nB) out[bm * 2 + n] = acc[r];
        }
    }
}

extern "C" void kernel_launch(void* const* d_in, const int* in_sizes, int n_in,
                              void* d_out, int out_size, void* d_ws, size_t ws_size,
                              hipStream_t stream) {
    const int*   x       = (const int*)  d_in[0];
    const float* emb     = (const float*)d_in[1];
    const float* W1_rel  = (const float*)d_in[2];
    const float* b1      = (const float*)d_in[3];
    const float* W1_root = (const float*)d_in[4];
    const float* W2_rel  = (const float*)d_in[5];
    const float* b2      = (const float*)d_in[6];
    const float* W2_root = (const float*)d_in[7];
    const float* fc_W    = (const float*)d_in[8];
    const float* fc_b    = (const float*)d_in[9];
    float* out = (float*)d_out;
    float* Mc  = (float*)d_ws;                 // 258 floats used

    const int nB = in_sizes[0] / 2;            // 65536

    gnn_fold_kernel<<<1, 256, 0, stream>>>(W1_rel, b1, W1_root,
                                           W2_rel, b2, W2_root,
                                           fc_W, fc_b, Mc);

    const int blocks = (nB + 127) / 128;       // 8 waves/block, 16 batches/wave
    gnn_main_kernel<<<blocks, 256, 0, stream>>>(x, emb, Mc, out, nB);
}